// TwoHotEmbedding_11072425689873
// MI455X (gfx1250) — compile-verified
//
#include <hip/hip_runtime.h>

// Two-hot embedding gather for MI455X (gfx1250, wave32).
//   out[token, :] = W[i1[token], :] + (i1==i2 ? 0 : W[i2[token], :])
// Memory-bound: ~64MB NT-stream write + ~103MB weight gather (L2-resident on
// the 192MB L2). 128-bit coalesced loads, non-temporal 128-bit stores.

typedef float v4f __attribute__((ext_vector_type(4)));

#define EMB 512
#define LANES_PER_TOKEN (EMB / 4)   // 128 lanes, each handling one float4
#define BLOCK 256                   // 8 wave32s; 2 tokens per block

__global__ __launch_bounds__(BLOCK) void twohot_embed_kernel(
    const int* __restrict__ idx1,
    const int* __restrict__ idx2,
    const float* __restrict__ weight,
    float* __restrict__ out,
    int n_tokens)
{
    const int tid = blockIdx.x * BLOCK + threadIdx.x;

    // Within a wave32, (tid >> 7) is constant (128 lanes/token, 32 lanes/wave)
    // -> force it scalar so index loads become s_load and row gathers become
    // SGPR-base + lane-offset global_load_b128.
    const int token = __builtin_amdgcn_readfirstlane(tid >> 7);
    if (token >= n_tokens) return;

    const int chunk = (tid & (LANES_PER_TOKEN - 1)) << 2;  // float offset in row

    const int i1 = __builtin_amdgcn_readfirstlane(idx1[token]);
    const int i2 = __builtin_amdgcn_readfirstlane(idx2[token]);

    // Both gathers issued unconditionally -> two b128 loads in flight.
    const v4f a = *(const v4f*)(weight + (size_t)i1 * EMB + chunk);
    const v4f b = *(const v4f*)(weight + (size_t)i2 * EMB + chunk);

    // scatter_(value=1) semantics: second row suppressed when indices match.
    const float m = (i1 == i2) ? 0.0f : 1.0f;  // one v_cndmask per wave-uniform pair

    v4f r;
    r.x = __builtin_fmaf(m, b.x, a.x);
    r.y = __builtin_fmaf(m, b.y, a.y);
    r.z = __builtin_fmaf(m, b.z, a.z);
    r.w = __builtin_fmaf(m, b.w, a.w);

    // Output is write-once streaming data: non-temporal store (th:TH_STORE_NT)
    // so the 64MB result doesn't evict the 103MB weight table from the 192MB L2.
    __builtin_nontemporal_store(r, (v4f*)(out + (size_t)token * EMB + chunk));
}

extern "C" void kernel_launch(void* const* d_in, const int* in_sizes, int n_in,
                              void* d_out, int out_size, void* d_ws, size_t ws_size,
                              hipStream_t stream) {
    (void)n_in; (void)out_size; (void)d_ws; (void)ws_size;

    const int*   idx1   = (const int*)d_in[0];    // input_one [B,S] int32
    const int*   idx2   = (const int*)d_in[1];    // input_two [B,S] int32
    const float* weight = (const float*)d_in[2];  // [VOCAB, 512] float32
    float*       out    = (float*)d_out;          // [B,S,512] float32

    const int n_tokens = in_sizes[0];                       // 8*4096 = 32768
    const long total_threads = (long)n_tokens * LANES_PER_TOKEN;
    const int n_blocks = (int)((total_threads + BLOCK - 1) / BLOCK);

    twohot_embed_kernel<<<n_blocks, BLOCK, 0, stream>>>(idx1, idx2, weight, out, n_tokens);
}